// SignedGCN_18837726560927
// MI455X (gfx1250) — compile-verified
//
#include <hip/hip_runtime.h>
#include <math.h>

#define NN 50000
#define EE 800000

typedef float v2f __attribute__((ext_vector_type(2)));
typedef float v8f __attribute__((ext_vector_type(8)));

// ---------------- utility kernels ----------------

__global__ void zero_kernel(float* __restrict__ p, int n) {
    int i = blockIdx.x * blockDim.x + threadIdx.x;
    if (i < n) p[i] = 0.0f;
}

__global__ void degree_kernel(const int* __restrict__ pd, const int* __restrict__ nd,
                              float* __restrict__ degP, float* __restrict__ degN) {
    int e = blockIdx.x * blockDim.x + threadIdx.x;
    if (e >= EE) return;
    atomicAdd(&degP[pd[e]], 1.0f);
    atomicAdd(&degN[nd[e]], 1.0f);
}

__global__ void invdeg_kernel(const float* __restrict__ degP, const float* __restrict__ degN,
                              float* __restrict__ invP, float* __restrict__ invN) {
    int i = blockIdx.x * blockDim.x + threadIdx.x;
    if (i >= NN) return;
    invP[i] = 1.0f / fmaxf(degP[i], 1.0f);
    invN[i] = 1.0f / fmaxf(degN[i], 1.0f);
}

// layer-1 scatter: 64-wide features from x
__global__ void scatter64_kernel(const float* __restrict__ x,
                                 const int* __restrict__ ps, const int* __restrict__ pd,
                                 const int* __restrict__ ns, const int* __restrict__ nd,
                                 float* __restrict__ aggP, float* __restrict__ aggN) {
    long long idx = (long long)blockIdx.x * blockDim.x + threadIdx.x;
    if (idx >= (long long)EE * 64) return;
    int e = (int)(idx >> 6);
    int f = (int)(idx & 63);
    atomicAdd(&aggP[(long long)pd[e] * 64 + f], x[(long long)ps[e] * 64 + f]);
    atomicAdd(&aggN[(long long)nd[e] * 64 + f], x[(long long)ns[e] * 64 + f]);
}

// later layers scatter: 128-wide features from z
__global__ void scatter128_kernel(const float* __restrict__ z,
                                  const int* __restrict__ ps, const int* __restrict__ pd,
                                  const int* __restrict__ ns, const int* __restrict__ nd,
                                  float* __restrict__ aggP, float* __restrict__ aggN) {
    long long idx = (long long)blockIdx.x * blockDim.x + threadIdx.x;
    if (idx >= (long long)EE * 128) return;
    int e = (int)(idx >> 7);
    int f = (int)(idx & 127);
    atomicAdd(&aggP[(long long)pd[e] * 128 + f], z[(long long)ps[e] * 128 + f]);
    atomicAdd(&aggN[(long long)nd[e] * 128 + f], z[(long long)ns[e] * 128 + f]);
}

// ---------------- WMMA GEMM kernels (f32 16x16x4) ----------------
// Block: 256 threads = 8 wave32s. Block b owns rows [16b,16b+16).
// Wave w<4: pos side, col tile w. Wave w>=4: neg side, col tile w-4.
// A-fragment layout (16x4 f32): lanes 0-15 hold (row=m0+lane, K=k..k+1),
// lanes 16-31 hold (row=m0+lane-16, K=k+2..k+3).
// B-fragment (4x16): lanes 0-15 hold col n0+lane rows k,k+1; lanes 16-31 rows k+2,k+3.
// C/D layout: VGPR r: lanes0-15 -> row m0+r, lanes16-31 -> row m0+8+r.

__global__ __launch_bounds__(256) void gemm1_kernel(
    const float* __restrict__ aggP, const float* __restrict__ aggN,
    const float* __restrict__ invP, const float* __restrict__ invN,
    const float* __restrict__ x,
    const float* __restrict__ Wp, const float* __restrict__ bp,
    const float* __restrict__ Wn, const float* __restrict__ bn,
    float* __restrict__ out) {
    int m0   = blockIdx.x * 16;
    int wave = threadIdx.x >> 5;
    int lane = threadIdx.x & 31;
    int sign = wave >> 2;           // 0 = pos, 1 = neg
    int n0   = (wave & 3) * 16;
    int lr   = lane & 15;
    int lh   = lane >> 4;

    const float* agg  = sign ? aggN : aggP;
    const float* inv  = sign ? invN : invP;
    const float* W    = sign ? Wn : Wp;
    const float* bias = sign ? bn : bp;

    int rowA = m0 + lr;
    float invv = inv[rowA];
    int col = n0 + lr;

    v8f acc = {};
    for (int k = 0; k < 128; k += 4) {
        int ca = k + lh * 2;        // this lane's two K indices: ca, ca+1
        v2f a;
        if (ca < 64) {
            const float2 t = *reinterpret_cast<const float2*>(agg + (long long)rowA * 64 + ca);
            a[0] = t.x * invv; a[1] = t.y * invv;
        } else {
            const float2 t = *reinterpret_cast<const float2*>(x + (long long)rowA * 64 + (ca - 64));
            a[0] = t.x; a[1] = t.y;
        }
        v2f b;
        b[0] = W[ca * 64 + col];
        b[1] = W[(ca + 1) * 64 + col];
        acc = __builtin_amdgcn_wmma_f32_16x16x4_f32(false, a, false, b, (short)0, acc, false, false);
    }
    float bb = bias[col];
    int colOut = sign * 64 + col;
#pragma unroll
    for (int r = 0; r < 8; ++r) {
        int row = m0 + r + lh * 8;
        out[(long long)row * 128 + colOut] = tanhf(acc[r] + bb);
    }
}

__device__ inline v2f loadA_L(const float* __restrict__ aggP, const float* __restrict__ aggN,
                              const float* __restrict__ zin,
                              float invPv, float invNv, int row, int c, int sign) {
    const float* base; long long off; float scale;
    if (c < 64) {            // pos: aggP[:, c] ; neg: aggP[:, 64+c]     (mean over pos edges)
        base = aggP; off = (long long)row * 128 + (sign ? 64 + c : c); scale = invPv;
    } else if (c < 128) {    // pos: aggN[:, c] ; neg: aggN[:, c-64]     (mean over neg edges)
        base = aggN; off = (long long)row * 128 + (sign ? c - 64 : c); scale = invNv;
    } else {                 // self features: zp for pos, zn for neg
        base = zin; off = (long long)row * 128 + (sign ? c - 64 : c - 128); scale = 1.0f;
    }
    const float2 t = *reinterpret_cast<const float2*>(base + off);
    v2f a; a[0] = t.x * scale; a[1] = t.y * scale;
    return a;
}

__global__ __launch_bounds__(256) void gemmL_kernel(
    const float* __restrict__ aggP, const float* __restrict__ aggN,
    const float* __restrict__ invP, const float* __restrict__ invN,
    const float* __restrict__ zin,
    const float* __restrict__ Wp, const float* __restrict__ bp,
    const float* __restrict__ Wn, const float* __restrict__ bn,
    float* __restrict__ out) {
    int m0   = blockIdx.x * 16;
    int wave = threadIdx.x >> 5;
    int lane = threadIdx.x & 31;
    int sign = wave >> 2;
    int n0   = (wave & 3) * 16;
    int lr   = lane & 15;
    int lh   = lane >> 4;

    const float* W    = sign ? Wn : Wp;
    const float* bias = sign ? bn : bp;

    int rowA = m0 + lr;
    float invPv = invP[rowA];
    float invNv = invN[rowA];
    int col = n0 + lr;

    v8f acc = {};
    for (int k = 0; k < 192; k += 4) {
        int ca = k + lh * 2;
        v2f a = loadA_L(aggP, aggN, zin, invPv, invNv, rowA, ca, sign);
        v2f b;
        b[0] = W[ca * 64 + col];
        b[1] = W[(ca + 1) * 64 + col];
        acc = __builtin_amdgcn_wmma_f32_16x16x4_f32(false, a, false, b, (short)0, acc, false, false);
    }
    float bb = bias[col];
    int colOut = sign * 64 + col;
#pragma unroll
    for (int r = 0; r < 8; ++r) {
        int row = m0 + r + lh * 8;
        out[(long long)row * 128 + colOut] = tanhf(acc[r] + bb);
    }
}

// ---------------- host-side orchestration ----------------

extern "C" void kernel_launch(void* const* d_in, const int* in_sizes, int n_in,
                              void* d_out, int out_size, void* d_ws, size_t ws_size,
                              hipStream_t stream) {
    (void)in_sizes; (void)n_in; (void)out_size; (void)ws_size;

    const float* x    = (const float*)d_in[0];
    const int*   pei  = (const int*)d_in[1];   // (2,E): [0:E]=src, [E:2E]=dst
    const int*   nei  = (const int*)d_in[2];
    const float* W1p  = (const float*)d_in[3];
    const float* b1p  = (const float*)d_in[4];
    const float* W1n  = (const float*)d_in[5];
    const float* b1n  = (const float*)d_in[6];
    const float* Wlp  = (const float*)d_in[7]; // (2,192,64)
    const float* blp  = (const float*)d_in[8]; // (2,64)
    const float* Wln  = (const float*)d_in[9];
    const float* bln  = (const float*)d_in[10];

    const int* ps = pei;       const int* pd = pei + EE;
    const int* ns = nei;       const int* nd = nei + EE;

    float* ws   = (float*)d_ws;
    float* degP = ws;                       // N
    float* degN = degP + NN;                // N
    float* invP = degN + NN;                // N
    float* invN = invP + NN;                // N
    float* aggP = invN + NN;                // N*128 (N*64 used in layer 1)
    float* aggN = aggP + (size_t)NN * 128;  // N*128
    float* zA   = aggN + (size_t)NN * 128;  // N*128
    float* zB   = zA   + (size_t)NN * 128;  // N*128
    float* zOut = (float*)d_out;            // N*128

    const int B = 256;
    const int rowTiles = NN / 16;           // 3125

    // degrees + reciprocals
    zero_kernel<<<(2 * NN + B - 1) / B, B, 0, stream>>>(degP, 2 * NN);
    degree_kernel<<<(EE + B - 1) / B, B, 0, stream>>>(pd, nd, degP, degN);
    invdeg_kernel<<<(NN + B - 1) / B, B, 0, stream>>>(degP, degN, invP, invN);

    // ---- layer 1: aggregate x (64-wide), GEMM K=128, tanh -> zA ----
    zero_kernel<<<(2 * NN * 64 + B - 1) / B, B, 0, stream>>>(aggP, 2 * NN * 64);
    {
        long long tot = (long long)EE * 64;
        scatter64_kernel<<<(unsigned)((tot + B - 1) / B), B, 0, stream>>>(
            x, ps, pd, ns, nd, aggP, aggP + (size_t)NN * 64);
        gemm1_kernel<<<rowTiles, B, 0, stream>>>(
            aggP, aggP + (size_t)NN * 64, invP, invN, x, W1p, b1p, W1n, b1n, zA);
    }

    // ---- layers 2..3: aggregate z (128-wide), GEMM K=192, tanh ----
    const float* zin = zA;
    for (int l = 0; l < 2; ++l) {
        float* zout = (l == 0) ? zB : zOut;
        zero_kernel<<<(2 * NN * 128 + B - 1) / B, B, 0, stream>>>(aggP, 2 * NN * 128);
        long long tot = (long long)EE * 128;
        scatter128_kernel<<<(unsigned)((tot + B - 1) / B), B, 0, stream>>>(
            zin, ps, pd, ns, nd, aggP, aggN);
        gemmL_kernel<<<rowTiles, B, 0, stream>>>(
            aggP, aggN, invP, invN, zin,
            Wlp + (size_t)l * 192 * 64, blp + (size_t)l * 64,
            Wln + (size_t)l * 192 * 64, bln + (size_t)l * 64,
            zout);
        zin = zout;
    }
}